// FlareEvolveGCN_22522808500496
// MI455X (gfx1250) — compile-verified
//
#include <hip/hip_runtime.h>
#include <hip/hip_bf16.h>
#include <math.h>

#define TSTEPS 3
#define DD 64
#define LP 65   // padded LDS row stride (floats) to avoid bank conflicts

typedef __attribute__((ext_vector_type(2))) float v2f;
typedef __attribute__((ext_vector_type(8))) float v8f;

__device__ __forceinline__ float sigf(float x) { return 1.0f / (1.0f + __expf(-x)); }

// ---------------------------------------------------------------------------
// Zero per-timestep scratch: mean[64] column sums and deg[N] counters.
// ---------------------------------------------------------------------------
__global__ void zero_kernel(float* __restrict__ mean, unsigned* __restrict__ deg, int N) {
    int i = blockIdx.x * blockDim.x + threadIdx.x;
    if (i < 64) mean[i] = 0.0f;
    if (i < N)  deg[i] = 0u;
}

// ---------------------------------------------------------------------------
// Column sums of x_t [N,64] -> mean[64] (sums; divided by N later).
// Coalesced: tid%64 = column, 4 rows per block-iteration.
// ---------------------------------------------------------------------------
__global__ void colsum_kernel(const float* __restrict__ x, float* __restrict__ mean, int N) {
    __shared__ float s[64];
    int tid = threadIdx.x;
    if (tid < 64) s[tid] = 0.0f;
    __syncthreads();
    int col   = tid & 63;
    int rpb   = blockDim.x >> 6;
    int row   = blockIdx.x * rpb + (tid >> 6);
    int rstep = gridDim.x * rpb;
    float acc = 0.0f;
    for (; row < N; row += rstep) acc += x[(size_t)row * 64 + col];
    atomicAdd(&s[col], acc);
    __syncthreads();
    if (tid < 64) unsafeAtomicAdd(&mean[tid], s[tid]);
}

// ---------------------------------------------------------------------------
// deg[d] = # edges with dst == d
// ---------------------------------------------------------------------------
__global__ void deg_kernel(const int* __restrict__ dst, unsigned* __restrict__ deg, int E) {
    int e = blockIdx.x * blockDim.x + threadIdx.x;
    if (e < E) atomicAdd(&deg[dst[e]], 1u);
}

// ---------------------------------------------------------------------------
// MatGRU weight evolution, one block of 512 threads (16 wave32 waves).
// Uses V_WMMA_F32_16X16X4_F32 for the three 64x64x64 fp32 GEMMs:
//   Zp = w @ Wz_bot, Rp = w @ Wr_bot, Hp = (r*w) @ Wh_bot
// Rank-1 parts (ctx broadcast rows) folded in as per-column bias vectors.
// Also emits v = w_new @ Wo (64-vector) for the fused edge pass.
// ---------------------------------------------------------------------------
__global__ __launch_bounds__(512) void gru_kernel(
    const float* __restrict__ w_in, const float* __restrict__ mean_sum,
    const float* __restrict__ Wz, const float* __restrict__ bz,
    const float* __restrict__ Wr, const float* __restrict__ br,
    const float* __restrict__ Wh, const float* __restrict__ bh,
    const float* __restrict__ Wp, const float* __restrict__ bp,
    const float* __restrict__ Wo,
    float* __restrict__ w_out, float* __restrict__ v_out,
    float* __restrict__ w_final, float invN)
{
    __shared__ float s_w [64 * LP];
    __shared__ float s_rw[64 * LP];
    __shared__ float s_z [64 * LP];
    __shared__ float s_wn[64 * LP];
    __shared__ float s_ctx[64], s_uz[64], s_ur[64], s_uh[64];

    int tid = threadIdx.x;

    for (int i = tid; i < 64 * 64; i += blockDim.x)
        s_w[(i >> 6) * LP + (i & 63)] = w_in[i];

    if (tid < 64) {                                   // ctx = mean @ Wp + bp
        float acc = bp[tid];
        for (int k = 0; k < 64; ++k) acc += (mean_sum[k] * invN) * Wp[k * 64 + tid];
        s_ctx[tid] = acc;
    }
    __syncthreads();
    if (tid < 64) {                                   // u* = ctx @ W*_top + b*
        float az = bz[tid], ar = br[tid], ah = bh[tid];
        for (int k = 0; k < 64; ++k) {
            float c = s_ctx[k];
            az += c * Wz[k * 64 + tid];
            ar += c * Wr[k * 64 + tid];
            ah += c * Wh[k * 64 + tid];
        }
        s_uz[tid] = az; s_ur[tid] = ar; s_uh[tid] = ah;
    }
    __syncthreads();

    int wave = tid >> 5, lane = tid & 31;
    int mi = (wave >> 2) * 16, ni = (wave & 3) * 16;
    int lm = lane & 15;
    int khalf = (lane >> 4) * 2;          // A/B VGPR K split across lane halves
    int mhalf = (lane >> 4) * 8;          // C/D M split across lane halves

    // Stage 1: Zp and Rp via fp32 WMMA, K in steps of 4
    v8f zacc = {0,0,0,0,0,0,0,0};
    v8f racc = {0,0,0,0,0,0,0,0};
    for (int k0 = 0; k0 < 64; k0 += 4) {
        int ka = k0 + khalf;
        v2f a;  a.x = s_w[(mi + lm) * LP + ka];     a.y = s_w[(mi + lm) * LP + ka + 1];
        v2f bzf; bzf.x = Wz[(64 + ka) * 64 + ni + lm]; bzf.y = Wz[(64 + ka + 1) * 64 + ni + lm];
        v2f brf; brf.x = Wr[(64 + ka) * 64 + ni + lm]; brf.y = Wr[(64 + ka + 1) * 64 + ni + lm];
        zacc = __builtin_amdgcn_wmma_f32_16x16x4_f32(false, a, false, bzf, (short)0, zacc, false, false);
        racc = __builtin_amdgcn_wmma_f32_16x16x4_f32(false, a, false, brf, (short)0, racc, false, false);
    }
    {
        int n = ni + lm;
        for (int rr = 0; rr < 8; ++rr) {
            int m = mi + mhalf + rr;
            float zv = sigf(zacc[rr] + s_uz[n]);
            float rv = sigf(racc[rr] + s_ur[n]);
            s_z [m * LP + n] = zv;
            s_rw[m * LP + n] = rv * s_w[m * LP + n];
        }
    }
    __syncthreads();

    // Stage 2: Hp = (r*w) @ Wh_bot
    v8f hacc = {0,0,0,0,0,0,0,0};
    for (int k0 = 0; k0 < 64; k0 += 4) {
        int ka = k0 + khalf;
        v2f a;  a.x = s_rw[(mi + lm) * LP + ka];  a.y = s_rw[(mi + lm) * LP + ka + 1];
        v2f bf; bf.x = Wh[(64 + ka) * 64 + ni + lm]; bf.y = Wh[(64 + ka + 1) * 64 + ni + lm];
        hacc = __builtin_amdgcn_wmma_f32_16x16x4_f32(false, a, false, bf, (short)0, hacc, false, false);
    }
    {
        int n = ni + lm;
        for (int rr = 0; rr < 8; ++rr) {
            int m = mi + mhalf + rr;
            float ht = tanhf(hacc[rr] + s_uh[n]);
            float zv = s_z[m * LP + n];
            float wn = zv * s_w[m * LP + n] + (1.0f - zv) * ht;
            s_wn[m * LP + n] = wn;
            w_out[m * 64 + n] = wn;
            if (w_final) w_final[m * 64 + n] = wn;
        }
    }
    __syncthreads();

    if (tid < 64) {                                   // v = w_new @ Wo
        float acc = 0.0f;
        for (int n = 0; n < 64; ++n) acc += s_wn[tid * LP + n] * Wo[n];
        v_out[tid] = acc;
    }
}

// ---------------------------------------------------------------------------
// y[n] = x_t[n,:] . v ; out_t[n] = bo   (fused out init)
// ---------------------------------------------------------------------------
__global__ void y_kernel(const float* __restrict__ x, const float* __restrict__ v,
                         float* __restrict__ y, float* __restrict__ out,
                         const float* __restrict__ bo, int N)
{
    __shared__ float sv[64];
    if (threadIdx.x < 64) sv[threadIdx.x] = v[threadIdx.x];
    __syncthreads();
    int n = blockIdx.x * blockDim.x + threadIdx.x;
    if (n >= N) return;
    const float4* xr = (const float4*)(x + (size_t)n * 64);
    float acc = 0.0f;
#pragma unroll
    for (int i = 0; i < 16; ++i) {
        float4 p = xr[i];
        acc += p.x * sv[4*i] + p.y * sv[4*i+1] + p.z * sv[4*i+2] + p.w * sv[4*i+3];
    }
    y[n]   = acc;
    out[n] = bo[0];
}

// ---------------------------------------------------------------------------
// out_t[dst_e] += y[src_e] / max(deg[dst_e], 1)
// ---------------------------------------------------------------------------
__global__ void edge_kernel(const int* __restrict__ src, const int* __restrict__ dst,
                            const float* __restrict__ y, const unsigned* __restrict__ deg,
                            float* __restrict__ out, int E)
{
    int e = blockIdx.x * blockDim.x + threadIdx.x;
    if (e >= E) return;
    int d = dst[e];
    unsigned dg = deg[d];
    float nrm = 1.0f / (float)(dg ? dg : 1u);
    unsafeAtomicAdd(&out[d], y[src[e]] * nrm);
}

// ---------------------------------------------------------------------------
extern "C" void kernel_launch(void* const* d_in, const int* in_sizes, int n_in,
                              void* d_out, int out_size, void* d_ws, size_t ws_size,
                              hipStream_t stream)
{
    const float* x    = (const float*)d_in[0];
    const int*   src  = (const int*)  d_in[1];
    const int*   dst  = (const int*)  d_in[2];
    const float* w0in = (const float*)d_in[3];
    const float* Wz = (const float*)d_in[4];  const float* bz = (const float*)d_in[5];
    const float* Wr = (const float*)d_in[6];  const float* br = (const float*)d_in[7];
    const float* Wh = (const float*)d_in[8];  const float* bh = (const float*)d_in[9];
    const float* Wp = (const float*)d_in[10]; const float* bp = (const float*)d_in[11];
    const float* Wo = (const float*)d_in[12]; const float* bo = (const float*)d_in[13];
    float* out = (float*)d_out;

    const int N = in_sizes[0] / (TSTEPS * 64);
    const int E = in_sizes[1] / TSTEPS;

    float* ws    = (float*)d_ws;
    float* mean  = ws;                       // 64
    float* v     = ws + 64;                  // 64
    float* wbuf0 = ws + 128;                 // 4096
    float* wbuf1 = ws + 128 + 4096;          // 4096
    unsigned* deg = (unsigned*)(ws + 128 + 8192);  // N u32
    float* y      = ws + 128 + 8192 + N;           // N f32

    const float invN = 1.0f / (float)N;
    const int zgrid = (N + 255) / 256;
    const int egrid = (E + 255) / 256;
    const int ngrid = (N + 255) / 256;

    for (int t = 0; t < TSTEPS; ++t) {
        const float* xt   = x   + (size_t)t * N * 64;
        const int*   st   = src + (size_t)t * E;
        const int*   dt   = dst + (size_t)t * E;
        float*       outt = out + (size_t)t * N;
        const float* wi = (t == 0) ? w0in : ((t == 1) ? wbuf0 : wbuf1);
        float*       wo = (t == 1) ? wbuf1 : wbuf0;
        float* wfin = (t == TSTEPS - 1) ? (out + (size_t)TSTEPS * N) : nullptr;

        zero_kernel  <<<zgrid, 256, 0, stream>>>(mean, deg, N);
        colsum_kernel<<<512,   256, 0, stream>>>(xt, mean, N);
        deg_kernel   <<<egrid, 256, 0, stream>>>(dt, deg, E);
        gru_kernel   <<<1,     512, 0, stream>>>(wi, mean, Wz, bz, Wr, br, Wh, bh,
                                                 Wp, bp, Wo, wo, v, wfin, invN);
        y_kernel     <<<ngrid, 256, 0, stream>>>(xt, v, y, outt, bo, N);
        edge_kernel  <<<egrid, 256, 0, stream>>>(st, dt, y, deg, outt, E);
    }
}